// Entangle_layer_4054449127786
// MI455X (gfx1250) — compile-verified
//
#include <hip/hip_runtime.h>

// CDNA5 / gfx1250: HBM-streaming kernel for the X/Z entangling layer.
// Traffic: 67 MB reads + 134 MB NT writes -> ~8.6 us floor @ 23.3 TB/s.
// WMMA intentionally unused (pointwise 2x2 complex gate, AI ~0.3 FLOP/B).
// This revision stages inputs through LDS with the gfx1250 async data mover:
//   global_load_async_to_lds_b128  (ASYNCcnt)  ->  s_wait_asynccnt 0  ->  ds_load_b128
// which is bandwidth-identical to direct b128 loads (512 B per wave-instruction).

typedef float v4f __attribute__((ext_vector_type(4)));

__global__ __launch_bounds__(256) void entangle_xz_kernel(
    const v4f* __restrict__ re4,
    const v4f* __restrict__ im4,
    v4f* __restrict__ out4,
    unsigned nvec)
{
    __shared__ v4f lds_re[256];
    __shared__ v4f lds_im[256];

    unsigned t = threadIdx.x;
    unsigned v = blockIdx.x * 256u + t;
    if (v >= nvec) return;

    // ---- CDNA5 async global->LDS staging (per-lane 16 B slot) ----
    // Low 32 bits of a generic LDS pointer are the LDS byte offset; the async
    // op takes its LDS byte address from the VDST VGPR.
    unsigned lre = (unsigned)(uintptr_t)&lds_re[t];
    unsigned lim = (unsigned)(uintptr_t)&lds_im[t];
    unsigned long long ga_re = (unsigned long long)(uintptr_t)(re4 + v);
    unsigned long long ga_im = (unsigned long long)(uintptr_t)(im4 + v);

    asm volatile("global_load_async_to_lds_b128 %0, %1, off th:TH_LOAD_NT"
                 :: "v"(lre), "v"(ga_re) : "memory");
    asm volatile("global_load_async_to_lds_b128 %0, %1, off th:TH_LOAD_NT"
                 :: "v"(lim), "v"(ga_im) : "memory");
    asm volatile("s_wait_asynccnt 0x0" ::: "memory");

    v4f x = lds_re[t];   // ds_load_b128
    v4f y = lds_im[t];   // ds_load_b128

    unsigned kv = v & 15u;          // float4 index inside K row (K=64 -> 16 vecs)
    unsigned a  = (v >> 4) & 1u;    // amplitude index of this qubit
    unsigned r  = (v >> 5) & 63u;   // pre-tile branch index
    unsigned bq = v >> 11;          // fused batch*qubit index, 0..1023

    const float INV = 0.84089641525371454f;   // 2^(-1/4)
    const float C0  = 0.59460355750136051f;   // 2^(-3/4)

    // ---- X branch: out[bq, r, 1-a, k] = 2^(-1/4) * z ----
    unsigned cbX = ((bq * 128u + r) * 2u + (1u - a)) * 64u + kv * 4u; // complex idx
    unsigned foX = cbX >> 1;                                          // float4 idx
    v4f o0, o1;
    o0.x = INV * x.x; o0.y = INV * y.x; o0.z = INV * x.y; o0.w = INV * y.y;
    o1.x = INV * x.z; o1.y = INV * y.z; o1.z = INV * x.w; o1.w = INV * y.w;
    __builtin_nontemporal_store(o0, out4 + foX);
    __builtin_nontemporal_store(o1, out4 + foX + 1);

    // ---- Z branch: out[bq, 64+r, a, k] = (a ? -1 : +1) * 2^(-3/4) * (1+i) * z
    //      (1+i)*(x+iy) = (x - y) + i*(x + y)
    float c = a ? -C0 : C0;
    unsigned cbZ = ((bq * 128u + 64u + r) * 2u + a) * 64u + kv * 4u;
    unsigned foZ = cbZ >> 1;
    v4f z0, z1;
    z0.x = c * (x.x - y.x); z0.y = c * (x.x + y.x);
    z0.z = c * (x.y - y.y); z0.w = c * (x.y + y.y);
    z1.x = c * (x.z - y.z); z1.y = c * (x.z + y.z);
    z1.z = c * (x.w - y.w); z1.w = c * (x.w + y.w);
    __builtin_nontemporal_store(z0, out4 + foZ);
    __builtin_nontemporal_store(z1, out4 + foZ + 1);
}

extern "C" void kernel_launch(void* const* d_in, const int* in_sizes, int n_in,
                              void* d_out, int out_size, void* d_ws, size_t ws_size,
                              hipStream_t stream) {
    (void)n_in; (void)out_size; (void)d_ws; (void)ws_size;
    const v4f* re4 = (const v4f*)d_in[0];     // state_re: (128,8,64,2,64) f32
    const v4f* im4 = (const v4f*)d_in[1];     // state_im: (128,8,64,2,64) f32
    v4f* out4 = (v4f*)d_out;                  // (128,8,128,2,64) complex64, interleaved

    unsigned nfloats = (unsigned)in_sizes[0]; // 8,388,608
    unsigned nvec = nfloats / 4u;             // 2,097,152 float4 lanes of work
    dim3 block(256);
    dim3 grid((nvec + 255u) / 256u);          // 8192 blocks = 65536 wave32 waves
    entangle_xz_kernel<<<grid, block, 0, stream>>>(re4, im4, out4, nvec);
}